// KAT_62079457296518
// MI455X (gfx1250) — compile-verified
//
#include <hip/hip_runtime.h>
#include <hip/hip_bf16.h>
#include <math.h>
#include <stdint.h>

// ---------------------------------------------------------------------------
// KAM-style dynamic-kernel conv pipeline for gfx1250 (MI455X).
// fp32 matrix math on V_WMMA_F32_16X16X4_F32 (exact fp32 FMA pipe),
// LDS-staged operands via the async global->LDS engine (ASYNCcnt),
// double-buffered staging overlapped with WMMA in the correlation GEMM.
// ---------------------------------------------------------------------------

typedef float v2f __attribute__((ext_vector_type(2)));
typedef float v8f __attribute__((ext_vector_type(8)));

static __device__ __forceinline__ v8f wmma_f32_16x16x4(v2f a, v2f b, v8f c) {
  // 8-arg pattern: (neg_a, A, neg_b, B, c_mod, C, reuse_a, reuse_b)
  return __builtin_amdgcn_wmma_f32_16x16x4_f32(false, a, false, b, (short)0, c,
                                               false, false);
}

// Async copy one dword global -> LDS (GLOBAL_LOAD_ASYNC_TO_LDS_B32, GV mode).
// lds_off: byte offset within the workgroup LDS allocation (generic __shared__
// pointers keep this in addr[31:0], ISA 10.2 aperture mapping).
static __device__ __forceinline__ void async_copy_f32(uint32_t lds_off,
                                                      const float* g) {
  asm volatile("global_load_async_to_lds_b32 %0, %1, off"
               :: "v"(lds_off), "v"((unsigned long long)(uintptr_t)g)
               : "memory");
}
static __device__ __forceinline__ void wait_async0() {
  asm volatile("s_wait_asynccnt 0x0" ::: "memory");
}
static __device__ __forceinline__ uint32_t lds_off32(const void* p) {
  return (uint32_t)(uintptr_t)p;
}

#define BATCH 16
#define CH    64
#define IMG   192
#define H2    64           // qk output spatial
#define QKC   128          // 2*CH
#define KK    576          // CH*3*3 reduction length
#define LSTR  68           // LDS row stride (floats) -> conflict-free column reads

// ---------------------------------------------------------------------------
// Kernel 1: qk = conv(x, qk_w, stride 3, VALID)  -> ws (B,128,64,64)
// Implicit GEMM: M=128 (oc), N=B*4096 (pixels), K=576. 16x16 tile per wave.
// grid = B * 8(Mtiles) * 256(Ntiles) = 32768 waves.
// ---------------------------------------------------------------------------
__global__ __launch_bounds__(32)
void qk_conv_wmma(const float* __restrict__ x, const float* __restrict__ qkw,
                  float* __restrict__ qk) {
  int id = blockIdx.x;
  int nt = id % 256; id /= 256;
  int mt = id % 8;   id /= 8;
  int b  = id;
  int py  = nt >> 2;           // row of 64
  int px0 = (nt & 3) * 16;     // 16-pixel segment
  int oc0 = mt * 16;

  int lane = threadIdx.x;
  int half = lane >> 4;        // K sub-offset selector
  int l    = lane & 15;

  int oc  = oc0 + l;           // A row (M)
  int npx = px0 + l;           // B col (N)

  const float* xb = x + (size_t)b * CH * IMG * IMG;
  v8f acc = {};

  for (int k0 = 0; k0 < KK; k0 += 4) {
    int kA = k0 + half * 2;
    int kB = kA + 1;
    int icA = kA / 9, rA = kA - icA * 9, kyA = rA / 3, kxA = rA - kyA * 3;
    int icB = kB / 9, rB = kB - icB * 9, kyB = rB / 3, kxB = rB - kyB * 3;

    v2f a, bm;
    a.x = qkw[((oc * CH + icA) * 3 + kyA) * 3 + kxA];
    a.y = qkw[((oc * CH + icB) * 3 + kyB) * 3 + kxB];
    // stride-3 VALID: coordinates always in bounds (3*63+2 = 191)
    bm.x = xb[((size_t)icA * IMG + (3 * py + kyA)) * IMG + (3 * npx + kxA)];
    bm.y = xb[((size_t)icB * IMG + (3 * py + kyB)) * IMG + (3 * npx + kxB)];
    acc = wmma_f32_16x16x4(a, bm, acc);
  }

#pragma unroll
  for (int r = 0; r < 8; ++r) {
    int m = oc0 + r + half * 8;
    qk[(((size_t)b * QKC + m) * H2 + py) * H2 + (px0 + l)] = acc[r];
  }
}

// ---------------------------------------------------------------------------
// Kernel 2: per-sample L2 norms of q (ch 0..63) and k (ch 64..127).
// grid = B*2 blocks of 256 threads.
// ---------------------------------------------------------------------------
__global__ void qk_norms(const float* __restrict__ qk, float* __restrict__ norms) {
  int b = blockIdx.x >> 1;
  int which = blockIdx.x & 1;
  const float* base = qk + ((size_t)b * QKC + which * CH) * (H2 * H2);
  float s = 0.f;
  for (int i = threadIdx.x; i < CH * H2 * H2; i += 256) {
    float v = base[i];
    s += v * v;
  }
  __shared__ float red[256];
  red[threadIdx.x] = s;
  __syncthreads();
  for (int off = 128; off > 0; off >>= 1) {
    if (threadIdx.x < off) red[threadIdx.x] += red[threadIdx.x + off];
    __syncthreads();
  }
  if (threadIdx.x == 0) norms[blockIdx.x] = sqrtf(red[0]);
}

// ---------------------------------------------------------------------------
// Kernel 3: xtx[b,co,ci,s] = scale_b * sum_{h,w} k[b,co,h,w]*q[b,ci,h+sy,w+sx]
// One 256-thread block (8 waves: 4 co-tiles x 2 ci-tiles) per (b, shift,
// ci-half). Row-slabs of k (64ch) and shifted q (32ch) are DOUBLE-BUFFERED in
// LDS and filled with global_load_async_to_lds_b32: while the 16 WMMAs for
// row h run, the async engine stages row h+1. Shift-window columns outside
// [w0,w1) are zeroed once up front, so the hot loop has no masking at all.
// grid = B*25*2 = 800 blocks.
// ---------------------------------------------------------------------------
__global__ __launch_bounds__(256)
void corr_wmma(const float* __restrict__ qk, const float* __restrict__ norms,
               float* __restrict__ xtx) {
  __shared__ float ks[2][CH * LSTR];        // 2 x 64ch x 68
  __shared__ float qs[2][32 * LSTR];        // 2 x 32ch x 68

  int part = blockIdx.x;
  int chalf = part % 2;  part /= 2;         // ci half (0 -> ci 0..31, 1 -> 32..63)
  int s = part % 25;
  int b = part / 25;
  int citBase = chalf * 32;

  int sy = s / 5 - 2, sx = s % 5 - 2;
  int h0 = (sy < 0) ? -sy : 0;
  int h1 = (sy > 0) ? (H2 - sy) : H2;
  int w0 = (sx < 0) ? -sx : 0;
  int w1 = (sx > 0) ? (H2 - sx) : H2;

  int tid  = threadIdx.x;
  int wave = tid >> 5;            // 0..7
  int cot  = wave >> 1;           // co tile (0..3)
  int citL = wave & 1;            // local ci tile (0..1)
  int lane = tid & 31, half = lane >> 4, l = lane & 15;

  int co  = cot * 16 + l;                    // A row (M) from k
  int ciL = citL * 16 + l;                   // local B col (N) from q

  const float* kbase = qk + ((size_t)b * QKC + CH) * (H2 * H2);  // k channels
  const float* qbase = qk + ((size_t)b * QKC) * (H2 * H2);       // q channels

  // ---- zero out-of-window columns once (async loads never touch them)
  for (int i = tid; i < CH * H2; i += 256) {
    int ch = i >> 6;
    int ww = i & 63;
    if (ww < w0 || ww >= w1) {
      ks[0][ch * LSTR + ww] = 0.f;
      ks[1][ch * LSTR + ww] = 0.f;
      if (ch < 32) {
        qs[0][ch * LSTR + ww] = 0.f;
        qs[1][ch * LSTR + ww] = 0.f;
      }
    }
  }

  // ---- async slab issue: k rows (64ch) + shifted q rows (32ch of this half)
  auto issue_slab = [&](int p, int hh) {
    for (int i = tid; i < CH * H2; i += 256) {
      int ch = i >> 6;
      int ww = i & 63;
      if (ww >= w0 && ww < w1) {
        async_copy_f32(lds_off32(&ks[p][ch * LSTR + ww]),
                       &kbase[(size_t)ch * (H2 * H2) + hh * H2 + ww]);
        if (ch < 32) {
          async_copy_f32(lds_off32(&qs[p][ch * LSTR + ww]),
                         &qbase[(size_t)(citBase + ch) * (H2 * H2) +
                                (hh + sy) * H2 + (ww + sx)]);
        }
      }
    }
  };

  v8f acc = {};

  issue_slab(0, h0);
  for (int h = h0; h < h1; ++h) {
    int p = (h - h0) & 1;
    wait_async0();          // this wave's slab for buffer p has landed
    __syncthreads();        // ... and everyone else's; prior reads of 1-p done
    if (h + 1 < h1) issue_slab(1 - p, h + 1);   // overlap next stage w/ WMMA

    // ---- 16 WMMAs across the 64-wide row, all operands from LDS, maskless
#pragma unroll
    for (int w = 0; w < H2; w += 4) {
      v2f a, bm;
      a.x  = ks[p][co * LSTR + w + half * 2];
      a.y  = ks[p][co * LSTR + w + half * 2 + 1];
      bm.x = qs[p][ciL * LSTR + w + half * 2];
      bm.y = qs[p][ciL * LSTR + w + half * 2 + 1];
      acc = wmma_f32_16x16x4(a, bm, acc);
    }
  }

  float nq = fmaxf(norms[b * 2 + 0], 1e-12f);
  float nk = fmaxf(norms[b * 2 + 1], 1e-12f);
  float scale = 1.f / (nq * nk);

#pragma unroll
  for (int r = 0; r < 8; ++r) {
    int com = cot * 16 + r + half * 8;
    int cin = citBase + citL * 16 + l;
    xtx[(((size_t)(b * CH + com) * CH + cin) * 25) + s] = acc[r] * scale;
  }
}

// ---------------------------------------------------------------------------
// Kernel 4: ak (permuted layout) [b][cw][co][ky][kx]
//   = sum_{ci,u,v} xtx[b,co,ci,ky+u,kx+v] * weight[cw,ci,u,v]
// 0.68 GFLOP -> direct VALU kernel. 589824 threads.
// ---------------------------------------------------------------------------
__global__ void ak_conv(const float* __restrict__ xtx, const float* __restrict__ weight,
                        float* __restrict__ akraw) {
  int idx = blockIdx.x * 256 + threadIdx.x;
  int kx = idx % 3;  int t = idx / 3;
  int ky = t % 3;    t /= 3;
  int co = t % CH;   t /= CH;
  int cw = t % CH;   t /= CH;
  int b  = t;
  const float* xb = xtx + ((size_t)(b * CH + co) * CH) * 25;
  const float* wb = weight + (size_t)cw * CH * 9;
  float sum = 0.f;
  for (int ci = 0; ci < CH; ++ci) {
    const float* xc = xb + ci * 25;
    const float* wc = wb + ci * 9;
#pragma unroll
    for (int u = 0; u < 3; ++u)
#pragma unroll
      for (int v = 0; v < 3; ++v)
        sum = fmaf(xc[(ky + u) * 5 + (kx + v)], wc[u * 3 + v], sum);
  }
  akraw[((((size_t)b * CH + cw) * CH + co) * 3 + ky) * 3 + kx] = sum;
}

// ---------------------------------------------------------------------------
// Kernel 5: per-(b,cw) L2 norm over 576 elems. grid = 1024 blocks of 64.
// ---------------------------------------------------------------------------
__global__ void ak_norms(const float* __restrict__ akraw, float* __restrict__ aknorm) {
  int bc = blockIdx.x;
  const float* base = akraw + (size_t)bc * KK;
  float s = 0.f;
  for (int i = threadIdx.x; i < KK; i += 64) {
    float v = base[i];
    s += v * v;
  }
  __shared__ float red[64];
  red[threadIdx.x] = s;
  __syncthreads();
  for (int off = 32; off > 0; off >>= 1) {
    if (threadIdx.x < off) red[threadIdx.x] += red[threadIdx.x + off];
    __syncthreads();
  }
  if (threadIdx.x == 0) aknorm[bc] = sqrtf(red[0]);
}

// ---------------------------------------------------------------------------
// Kernel 6: kern = weight + (ak/||ak||) * temperature   (elementwise, 589824)
// ---------------------------------------------------------------------------
__global__ void build_kern(const float* __restrict__ akraw, const float* __restrict__ aknorm,
                           const float* __restrict__ weight, const float* __restrict__ temp,
                           float* __restrict__ kern) {
  int idx = blockIdx.x * 256 + threadIdx.x;
  int within = idx % KK;        // (co,ky,kx)
  int bc = idx / KK;            // b*64+cw
  int cw = bc % CH;
  float n = fmaxf(aknorm[bc], 1e-12f);
  kern[idx] = fmaf(akraw[idx], temp[0] / n, weight[cw * KK + within]);
}

// ---------------------------------------------------------------------------
// Kernel 7 (dominant, 43.5 GFLOP): per-sample grouped 3x3 conv + relu.
// Implicit GEMM per b: M=64 (cw), N=36864 (pixels), K=576 (co,ky,kx), pad=1.
// One 128-thread block (4 waves) per (b, row h, 64-px segment): the block
// async-stages the x halo tile (64ch x 3rows x 66cols) straight into LDS
// (ds_store only for the zero border), then each wave computes a 16(M)x64(N)
// tile with all B-operands from LDS. kern A-loads are global (L2-resident).
// grid = B * 192 * 3 = 9216 blocks.
// ---------------------------------------------------------------------------
__global__ __launch_bounds__(128)
void main_conv_wmma(const float* __restrict__ x, const float* __restrict__ kern,
                    float* __restrict__ out) {
  __shared__ float xs[CH * 3 * LSTR];   // 64ch x 3rows x 68(stride), cols 0..65 used

  int id = blockIdx.x;
  int wt = id % 3;    id /= 3;
  int h  = id % IMG;  id /= IMG;
  int b  = id;
  int wbase = wt * 64;

  int tid  = threadIdx.x;
  int cwt  = tid >> 5;                  // wave -> output-channel tile
  int lane = tid & 31, half = lane >> 4, l = lane & 15;
  int cw   = cwt * 16 + l;

  const float* xb = x + (size_t)b * CH * IMG * IMG;
  const float* kb = kern + ((size_t)b * CH + cw) * KK;

  // warm L2 for this sample's kern slice (fire-and-forget prefetch)
  __builtin_prefetch(kern + ((size_t)b * CH) * KK + (size_t)tid * (CH * KK / 128), 0, 1);

  // ---- async-stage x halo tile: rows h-1..h+1, cols wbase-1..wbase+64
  for (int i = tid; i < CH * 3 * 66; i += 128) {
    int cc = i % 66;  int t2 = i / 66;
    int ky = t2 % 3;  int ic = t2 / 3;
    int hh = h + ky - 1;
    int ww = wbase - 1 + cc;
    int li = (ic * 3 + ky) * LSTR + cc;
    if (hh >= 0 && hh < IMG && ww >= 0 && ww < IMG)
      async_copy_f32(lds_off32(&xs[li]), &xb[((size_t)ic * IMG + hh) * IMG + ww]);
    else
      xs[li] = 0.f;                      // zero border via ds_store
  }
  wait_async0();
  __syncthreads();

  v8f acc[4] = {};

  for (int k0 = 0; k0 < KK; k0 += 4) {
    int kA = k0 + half * 2;
    int kB = kA + 1;
    int icA = kA / 9, rA = kA - icA * 9, kyA = rA / 3, kxA = rA - kyA * 3;
    int icB = kB / 9, rB = kB - icB * 9, kyB = rB / 3, kxB = rB - kyB * 3;

    v2f a;
    a.x = kb[kA];
    a.y = kb[kB];

    int baseA = (icA * 3 + kyA) * LSTR + l + kxA;
    int baseB = (icB * 3 + kyB) * LSTR + l + kxB;

#pragma unroll
    for (int t = 0; t < 4; ++t) {
      v2f bm;
      bm.x = xs[baseA + t * 16];
      bm.y = xs[baseB + t * 16];
      acc[t] = wmma_f32_16x16x4(a, bm, acc[t]);
    }
  }

#pragma unroll
  for (int t = 0; t < 4; ++t) {
#pragma unroll
    for (int r = 0; r < 8; ++r) {
      int m = cwt * 16 + r + half * 8;
      float v = acc[t][r];
      out[(((size_t)b * CH + m) * IMG + h) * IMG + (wbase + t * 16 + l)] =
          fmaxf(v, 0.f);
    }
  }
}

// ---------------------------------------------------------------------------
// Launch
// ---------------------------------------------------------------------------
extern "C" void kernel_launch(void* const* d_in, const int* in_sizes, int n_in,
                              void* d_out, int out_size, void* d_ws, size_t ws_size,
                              hipStream_t stream) {
  (void)in_sizes; (void)n_in; (void)out_size; (void)ws_size;

  const float* x      = (const float*)d_in[0];   // (16,64,192,192)
  const float* qk_w   = (const float*)d_in[1];   // (128,64,3,3)
  const float* weight = (const float*)d_in[2];   // (64,64,3,3)
  const float* temp   = (const float*)d_in[3];   // (1,1,1)
  float* out = (float*)d_out;                    // (16,64,192,192)

  float* ws     = (float*)d_ws;
  float* qk     = ws;                                   // 16*128*64*64 = 8388608
  float* norms  = qk + (size_t)BATCH * QKC * H2 * H2;   // 32 (pad to 64)
  float* xtx    = norms + 64;                           // 16*64*64*25 = 1638400
  float* akraw  = xtx + (size_t)BATCH * CH * CH * 25;   // 16*64*64*9  = 589824
  float* aknorm = akraw + (size_t)BATCH * CH * CH * 9;  // 1024
  float* kern   = aknorm + 1024;                        // 589824

  qk_conv_wmma  <<<BATCH * 8 * 256, 32, 0, stream>>>(x, qk_w, qk);
  qk_norms      <<<BATCH * 2, 256, 0, stream>>>(qk, norms);
  corr_wmma     <<<BATCH * 25 * 2, 256, 0, stream>>>(qk, norms, xtx);
  ak_conv       <<<(BATCH * CH * CH * 9) / 256, 256, 0, stream>>>(xtx, weight, akraw);
  ak_norms      <<<BATCH * CH, 64, 0, stream>>>(akraw, aknorm);
  build_kern    <<<(BATCH * CH * CH * 9) / 256, 256, 0, stream>>>(akraw, aknorm, weight, temp, kern);
  main_conv_wmma<<<BATCH * IMG * 3, 128, 0, stream>>>(x, kern, out);
}